// SpatialSelfCorrelation_78022375899283
// MI455X (gfx1250) — compile-verified
//
#include <hip/hip_runtime.h>
#include <math.h>

// ---------------------------------------------------------------------------
// SpatialSelfCorrelation for MI455X (gfx1250), fp32 WMMA flash-style attention.
//   x:    [64][8192] f32, W: [32][64] f32, bias: [32] f32, out: [64][8192] f32
// ---------------------------------------------------------------------------

typedef __attribute__((ext_vector_type(2))) float v2f;
typedef __attribute__((ext_vector_type(8))) float v8f;

#define N_TOK   8192
#define C_CH    64
#define CRDIM   32
#define SCALE   0.17677669529663687f   /* 1/sqrt(32) */
#define MCHUNKS 8
#define MCHUNK  (N_TOK / MCHUNKS)      /* 1024 */
#define PROW    17                     /* padded LDS row stride for P tile */

// D = A(16x4 f32) * B(4x16 f32) + C(16x16 f32); wave32, EXEC must be all 1s.
__device__ __forceinline__ v8f wmma4_f32(v2f a, v2f b, v8f c) {
    return __builtin_amdgcn_wmma_f32_16x16x4_f32(false, a, false, b,
                                                 (short)0, c, false, false);
}

// Load 8 K-step fragments of a 16-row q block. The same gather serves both the
// A-fragment (rows) and B-fragment (cols) of the Gram matrix S = Q^T Q:
//   lanes 0-15  hold K offsets {4k, 4k+1} of vector (row0 + lane)
//   lanes 16-31 hold K offsets {4k+2,4k+3} of vector (row0 + lane-16)
__device__ __forceinline__ void load_q_frags(const float* __restrict__ qT,
                                             int row0, int idx, int half,
                                             v2f* __restrict__ frag) {
    const float* p = qT + (size_t)(row0 + idx) * CRDIM + (half ? 2 : 0);
#pragma unroll
    for (int k = 0; k < 8; ++k)
        frag[k] = *(const v2f*)(p + 4 * k);
}

// --------------------------- kernel 1: projection ---------------------------
// qT[n][o] = bias[o] + sum_c W[o][c] * x[c][n]
__global__ void __launch_bounds__(128)
proj_kernel(const float* __restrict__ x, const float* __restrict__ W,
            const float* __restrict__ bias, float* __restrict__ qT) {
    int n = blockIdx.x * blockDim.x + threadIdx.x;
    if (n >= N_TOK) return;
    float acc[CRDIM];
#pragma unroll
    for (int o = 0; o < CRDIM; ++o) acc[o] = bias[o];
#pragma unroll 4
    for (int c = 0; c < C_CH; ++c) {
        float xv = x[(size_t)c * N_TOK + n];
#pragma unroll
        for (int o = 0; o < CRDIM; ++o) acc[o] += W[o * C_CH + c] * xv;
    }
#pragma unroll
    for (int o = 0; o < CRDIM; ++o) qT[(size_t)n * CRDIM + o] = acc[o];
}

// ----------------------- kernel 2: row softmax stats ------------------------
// One wave32 per (16-row block, 1024-col chunk). Online max/sum per lane, then
// width-16 shuffle reduction (wave32: halves hold rows r and r+8).
__global__ void __launch_bounds__(32)
stats_kernel(const float* __restrict__ qT, float* __restrict__ pmax,
             float* __restrict__ psum) {
    const int lane = threadIdx.x & 31;
    const int half = lane >> 4, idx = lane & 15;
    const int n0 = blockIdx.x * 16;
    const int m_begin = blockIdx.y * MCHUNK;

    v2f aq[8];
    load_q_frags(qT, n0, idx, half, aq);

    float mx[8], sm[8];
#pragma unroll
    for (int r = 0; r < 8; ++r) { mx[r] = -INFINITY; sm[r] = 0.0f; }

    for (int m0 = m_begin; m0 < m_begin + MCHUNK; m0 += 16) {
        v2f bq[8];
        load_q_frags(qT, m0, idx, half, bq);
        v8f s = {0, 0, 0, 0, 0, 0, 0, 0};
#pragma unroll
        for (int k = 0; k < 8; ++k) s = wmma4_f32(aq[k], bq[k], s);
#pragma unroll
        for (int r = 0; r < 8; ++r) {
            float v  = s[r] * SCALE;
            float nm = fmaxf(mx[r], v);
            sm[r] = sm[r] * __expf(mx[r] - nm) + __expf(v - nm);
            mx[r] = nm;
        }
    }
#pragma unroll
    for (int r = 0; r < 8; ++r) {
        float m = mx[r], s = sm[r];
#pragma unroll
        for (int off = 1; off < 16; off <<= 1) {
            float om = __shfl_xor(m, off, 16);
            float os = __shfl_xor(s, off, 16);
            float nm = fmaxf(m, om);
            s = s * __expf(m - nm) + os * __expf(om - nm);
            m = nm;
        }
        if (idx == 0) {
            int row = n0 + r + 8 * half;
            pmax[(size_t)blockIdx.y * N_TOK + row] = m;
            psum[(size_t)blockIdx.y * N_TOK + row] = s;
        }
    }
}

// -------------------- kernel 3: combine chunk statistics --------------------
__global__ void __launch_bounds__(128)
finalize_stats_kernel(const float* __restrict__ pmax,
                      const float* __restrict__ psum,
                      float* __restrict__ rowmax, float* __restrict__ rowinv) {
    int n = blockIdx.x * blockDim.x + threadIdx.x;
    if (n >= N_TOK) return;
    float M = -INFINITY;
#pragma unroll
    for (int j = 0; j < MCHUNKS; ++j) M = fmaxf(M, pmax[j * N_TOK + n]);
    float S = 0.0f;
#pragma unroll
    for (int j = 0; j < MCHUNKS; ++j)
        S += psum[j * N_TOK + n] * __expf(pmax[j * N_TOK + n] - M);
    rowmax[n] = M;
    rowinv[n] = 1.0f / S;
}

// ------------------- kernel 4: fused P recompute + V*P ---------------------
// One workgroup (4 waves) per 16-column m-block. Wave w handles n-blocks
// n0 = (w + 4t)*16; per n-block: 8 WMMA for S, LDS relayout of P (C-layout ->
// B-layout), then 16 WMMA accumulating out[0..63][m-block]. Cross-wave partials
// combined through LDS, residual x added on store.
__global__ void __launch_bounds__(128)
attn_out_kernel(const float* __restrict__ x, const float* __restrict__ qT,
                const float* __restrict__ rowmax,
                const float* __restrict__ rowinv, float* __restrict__ out) {
    __shared__ float Pbuf[4][16 * PROW];   // per-wave P tile (padded)
    __shared__ float Obuf[4][C_CH * 16];   // per-wave output partials

    const int tid  = threadIdx.x;
    const int w    = tid >> 5;
    const int lane = tid & 31;
    const int half = lane >> 4, idx = lane & 15;
    const int m0   = blockIdx.x * 16;

    v2f bq[8];                              // B-fragments of q for columns m0..m0+15
    load_q_frags(qT, m0, idx, half, bq);

    v8f acc[4];
#pragma unroll
    for (int ct = 0; ct < 4; ++ct) acc[ct] = (v8f){0, 0, 0, 0, 0, 0, 0, 0};

    float* P = &Pbuf[w][0];

    for (int t = 0; t < (N_TOK / 16) / 4; ++t) {
        const int n0 = (w + 4 * t) * 16;

        // --- S tile = q[n-block]^T q[m-block], K = 32 in 8 WMMA steps ---
        v2f aq[8];
        load_q_frags(qT, n0, idx, half, aq);
        v8f s = {0, 0, 0, 0, 0, 0, 0, 0};
#pragma unroll
        for (int k = 0; k < 8; ++k) s = wmma4_f32(aq[k], bq[k], s);

        // --- P = exp(s*scale - rowmax[n]) * rowinv[n]  ->  LDS [n][m] ---
#pragma unroll
        for (int r = 0; r < 8; ++r) {
            int row = n0 + r + 8 * half;
            float p = __expf(s[r] * SCALE - rowmax[row]) * rowinv[row];
            P[(r + 8 * half) * PROW + idx] = p;
        }
        // wave-local LDS RAW: compiler inserts s_wait_dscnt before the reads

        // --- out += V(16c x 4n) * P(4n x 16m), 4 K-steps x 4 c-tiles ---
#pragma unroll
        for (int kk = 0; kk < 4; ++kk) {
            const int nrow = 4 * kk + 2 * half;
            v2f bp;
            bp.x = P[nrow * PROW + idx];
            bp.y = P[(nrow + 1) * PROW + idx];
#pragma unroll
            for (int ct = 0; ct < 4; ++ct) {
                const float* vp = x + (size_t)(16 * ct + idx) * N_TOK
                                    + n0 + 4 * kk + 2 * half;
                v2f av = *(const v2f*)vp;
                acc[ct] = wmma4_f32(av, bp, acc[ct]);
            }
        }
    }

    // --- dump per-wave partials, combine across waves, add residual ---
#pragma unroll
    for (int ct = 0; ct < 4; ++ct)
#pragma unroll
        for (int r = 0; r < 8; ++r)
            Obuf[w][(16 * ct + r + 8 * half) * 16 + idx] = acc[ct][r];
    __syncthreads();

    for (int i = tid; i < C_CH * 16; i += 128) {
        int c = i >> 4, m = i & 15;
        float v = Obuf[0][i] + Obuf[1][i] + Obuf[2][i] + Obuf[3][i];
        out[(size_t)c * N_TOK + m0 + m] = v + x[(size_t)c * N_TOK + m0 + m];
    }
}

// ---------------------------------------------------------------------------
extern "C" void kernel_launch(void* const* d_in, const int* in_sizes, int n_in,
                              void* d_out, int out_size, void* d_ws,
                              size_t ws_size, hipStream_t stream) {
    const float* x    = (const float*)d_in[0];
    const float* W    = (const float*)d_in[1];
    const float* bias = (const float*)d_in[2];
    float* out = (float*)d_out;

    // workspace layout (floats): qT | pmax | psum | rowmax | rowinv  (~1.6 MB)
    float* ws     = (float*)d_ws;
    float* qT     = ws;                           // 8192*32
    float* pmax   = qT + (size_t)N_TOK * CRDIM;   // 8*8192
    float* psum   = pmax + (size_t)MCHUNKS * N_TOK;
    float* rowmax = psum + (size_t)MCHUNKS * N_TOK;
    float* rowinv = rowmax + N_TOK;

    proj_kernel<<<N_TOK / 128, 128, 0, stream>>>(x, W, bias, qT);
    stats_kernel<<<dim3(N_TOK / 16, MCHUNKS), 32, 0, stream>>>(qT, pmax, psum);
    finalize_stats_kernel<<<N_TOK / 128, 128, 0, stream>>>(pmax, psum, rowmax,
                                                           rowinv);
    attn_out_kernel<<<N_TOK / 16, 128, 0, stream>>>(x, qT, rowmax, rowinv, out);
}